// Net_48747878810173
// MI455X (gfx1250) — compile-verified
//
#include <hip/hip_runtime.h>
#include <math.h>

#define N_NODES  100000
#define N_EDGES  1600000
#define N_GRAPHS 256

typedef __attribute__((ext_vector_type(2))) float v2f;
typedef __attribute__((ext_vector_type(8))) float v8f;

// ---------------------------------------------------------------------------
// Degree count: cnt[dst] += 1 over all edges (dst is layer-invariant, do once)
// ---------------------------------------------------------------------------
__global__ void degree_kernel(const int* __restrict__ dst, float* __restrict__ cnt,
                              int n_edges) {
    int e = blockIdx.x * blockDim.x + threadIdx.x;
    if (e < n_edges) unsafeAtomicAdd(&cnt[dst[e]], 1.0f);
}

__global__ void invdeg_kernel(const float* __restrict__ cnt, float* __restrict__ inv,
                              int n) {
    int i = blockIdx.x * blockDim.x + threadIdx.x;
    if (i < n) inv[i] = 1.0f / fmaxf(cnt[i], 1.0f);
}

// ---------------------------------------------------------------------------
// Edge scatter: agg[dst] += h[src], float4 per thread (din % 4 == 0),
// blockIdx.y selects the feature chunk. Native f32 L2 atomics.
// ---------------------------------------------------------------------------
__global__ void sage_scatter(const float* __restrict__ h,
                             const int* __restrict__ src,
                             const int* __restrict__ dst,
                             float* __restrict__ agg,
                             int n_edges, int din) {
    int e = blockIdx.x * blockDim.x + threadIdx.x;
    if (e >= n_edges) return;
    int c = blockIdx.y << 2;
    int s = src[e];
    int d = dst[e];
    const float4 v = *reinterpret_cast<const float4*>(h + (size_t)s * din + c);
    float* p = agg + (size_t)d * din + c;
    unsafeAtomicAdd(p + 0, v.x);
    unsafeAtomicAdd(p + 1, v.y);
    unsafeAtomicAdd(p + 2, v.z);
    unsafeAtomicAdd(p + 3, v.w);
}

// ---------------------------------------------------------------------------
// Pack weights for WMMA: transpose [din x dout] -> [doutp x din] with the
// output dim zero-padded up to doutp (multiple of 16). Bias padded too.
// Removes ALL masking from the GEMM inner loop.
// ---------------------------------------------------------------------------
__global__ void pack_weights(const float* __restrict__ Wl,
                             const float* __restrict__ bl,
                             const float* __restrict__ Wr,
                             float* __restrict__ Wlt,
                             float* __restrict__ Wrt,
                             float* __restrict__ blp,
                             int din, int dout, int doutp) {
    int i = blockIdx.x * blockDim.x + threadIdx.x;
    int total = doutp * din;
    if (i < total) {
        int n = i / din;
        int k = i - n * din;
        bool ok = n < dout;
        Wlt[i] = ok ? Wl[k * dout + n] : 0.0f;
        Wrt[i] = ok ? Wr[k * dout + n] : 0.0f;
    }
    if (i < doutp) blp[i] = (i < dout) ? bl[i] : 0.0f;
}

// ---------------------------------------------------------------------------
// Fused SAGE linear: out = relu( (agg * invdeg) @ Wl + bl + h @ Wr )
// One wave32 per 16-node tile, V_WMMA_F32_16X16X4_F32, fully unrolled.
//
// A 16x4 f32 layout (ISA 7.12.2): lanes 0-15 hold rows M=lane with K0/K1 in
// the two A VGPRs; lanes 16-31 hold the same rows with K2/K3. B 4x16 mirrors
// it. C/D 16x16 f32: VGPR j = row j (lanes 0-15) / j+8 (lanes 16-31), N=l16.
//
// Weights live in LDS (transposed+padded -> contiguous ds_load_b64 fragments).
// Two independent accumulator chains (Wl-term / Wr-term) to break the
// WMMA->WMMA RAW hazard chain.
// ---------------------------------------------------------------------------
template <int DIN, int DOUT, int DOUTP>
__global__ void sage_fused_gemm(const float* __restrict__ h,
                                const float* __restrict__ agg,
                                const float* __restrict__ invdeg,
                                const float* __restrict__ Wlt,  // [DOUTP x DIN]
                                const float* __restrict__ blp,  // [DOUTP]
                                const float* __restrict__ Wrt,  // [DOUTP x DIN]
                                float* __restrict__ out,
                                int n_nodes) {
    constexpr int NT     = DOUTP / 16;
    constexpr int KSTEPS = DIN / 4;

    __shared__ float ldsW[2 * DOUTP * DIN + DOUTP];
    float* ldsWl = ldsW;
    float* ldsWr = ldsW + DOUTP * DIN;
    float* ldsb  = ldsW + 2 * DOUTP * DIN;
    for (int i = threadIdx.x; i < DOUTP * DIN; i += blockDim.x) {
        ldsWl[i] = Wlt[i];
        ldsWr[i] = Wrt[i];
    }
    for (int i = threadIdx.x; i < DOUTP; i += blockDim.x) ldsb[i] = blp[i];
    __syncthreads();

    const int lane = threadIdx.x & 31;
    const int wave = (blockIdx.x * blockDim.x + threadIdx.x) >> 5;
    const int m0   = wave << 4;
    if (m0 >= n_nodes) return;  // wave-uniform: EXEC stays all-ones for WMMA

    const int   half = lane >> 4;   // 0 -> K+0/1, rows 0-7 ; 1 -> K+2/3, rows 8-15
    const int   l16  = lane & 15;
    const int   row  = m0 + l16;
    const float rinv = invdeg[row];
    const float* hrow = h   + (size_t)row * DIN;
    const float* arow = agg + (size_t)row * DIN;

    // Preload A fragments once; reused for every 16-wide output tile.
    v2f aA[KSTEPS], aH[KSTEPS];
#pragma unroll
    for (int ks = 0; ks < KSTEPS; ++ks) {
        const int ka = ks * 4 + half * 2;
        aA[ks] = *reinterpret_cast<const v2f*>(arow + ka) * rinv;
        aH[ks] = *reinterpret_cast<const v2f*>(hrow + ka);
    }

#pragma unroll
    for (int t = 0; t < NT; ++t) {
        const int ncol = t * 16 + l16;
        const float* wlc = ldsWl + ncol * DIN;
        const float* wrc = ldsWr + ncol * DIN;
        v8f accL = {};
        v8f accR = {};
#pragma unroll
        for (int ks = 0; ks < KSTEPS; ++ks) {
            const int ka = ks * 4 + half * 2;
            v2f bL = *reinterpret_cast<const v2f*>(wlc + ka);
            v2f bR = *reinterpret_cast<const v2f*>(wrc + ka);
            accL = __builtin_amdgcn_wmma_f32_16x16x4_f32(
                false, aA[ks], false, bL, (short)0, accL, false, false);
            accR = __builtin_amdgcn_wmma_f32_16x16x4_f32(
                false, aH[ks], false, bR, (short)0, accR, false, false);
        }
        v8f acc = accL + accR;
        if (DOUT == DOUTP || ncol < DOUT) {   // only layer 3 (dout=2) masks
            const float b = ldsb[ncol];
#pragma unroll
            for (int j = 0; j < 8; ++j) {
                const int m = m0 + half * 8 + j;
                float v = acc[j] + b;
                out[(size_t)m * DOUT + ncol] = v > 0.0f ? v : 0.0f;
            }
        }
    }
}

// ---------------------------------------------------------------------------
// Global mean pool accumulation per graph (final h is N x 2)
// ---------------------------------------------------------------------------
__global__ void pool_kernel(const float* __restrict__ h,
                            const int* __restrict__ batch,
                            float* __restrict__ pooled,
                            float* __restrict__ gcnt, int n_nodes) {
    int n = blockIdx.x * blockDim.x + threadIdx.x;
    if (n >= n_nodes) return;
    int g = batch[n];
    unsafeAtomicAdd(&pooled[g * 2 + 0], h[(size_t)n * 2 + 0]);
    unsafeAtomicAdd(&pooled[g * 2 + 1], h[(size_t)n * 2 + 1]);
    unsafeAtomicAdd(&gcnt[g], 1.0f);
}

__global__ void softmax_kernel(const float* __restrict__ pooled,
                               const float* __restrict__ gcnt,
                               float* __restrict__ out, int n_graphs) {
    int g = blockIdx.x * blockDim.x + threadIdx.x;
    if (g >= n_graphs) return;
    float c  = fmaxf(gcnt[g], 1.0f);
    float p0 = pooled[g * 2 + 0] / c;
    float p1 = pooled[g * 2 + 1] / c;
    float m  = fmaxf(p0, p1);
    float e0 = expf(p0 - m);
    float e1 = expf(p1 - m);
    float s  = e0 + e1;
    out[g * 2 + 0] = e0 / s;
    out[g * 2 + 1] = e1 / s;
}

// ---------------------------------------------------------------------------
extern "C" void kernel_launch(void* const* d_in, const int* in_sizes, int n_in,
                              void* d_out, int out_size, void* d_ws, size_t ws_size,
                              hipStream_t stream) {
    const float* x     = (const float*)d_in[0];
    const int*   ei    = (const int*)d_in[1];      // [2, E] row-major
    const int*   batch = (const int*)d_in[2];
    const float* Wls[4] = {(const float*)d_in[3],  (const float*)d_in[6],
                           (const float*)d_in[9],  (const float*)d_in[12]};
    const float* bls[4] = {(const float*)d_in[4],  (const float*)d_in[7],
                           (const float*)d_in[10], (const float*)d_in[13]};
    const float* Wrs[4] = {(const float*)d_in[5],  (const float*)d_in[8],
                           (const float*)d_in[11], (const float*)d_in[14]};
    const int* src = ei;
    const int* dst = ei + N_EDGES;

    // Workspace carve (256B aligned): ~78 MB total, L2-resident
    size_t off  = 0;
    char*  base = (char*)d_ws;
    auto carve = [&](size_t bytes) -> float* {
        float* p = (float*)(base + off);
        off += (bytes + 255) & ~(size_t)255;
        return p;
    };
    float* cnt    = carve((size_t)N_NODES * 4);
    float* inv    = carve((size_t)N_NODES * 4);
    float* agg    = carve((size_t)N_NODES * 64 * 4);
    float* hA     = carve((size_t)N_NODES * 64 * 4);
    float* hB     = carve((size_t)N_NODES * 64 * 4);
    float* Wlt    = carve((size_t)4096 * 4);
    float* Wrt    = carve((size_t)4096 * 4);
    float* blp    = carve((size_t)64 * 4);
    float* pooled = carve((size_t)N_GRAPHS * 2 * 4);
    float* gcnt   = carve((size_t)N_GRAPHS * 4);
    (void)ws_size; (void)in_sizes; (void)n_in; (void)out_size;

    const int eblocks = (N_EDGES + 255) / 256;
    const int nblocks = (N_NODES + 255) / 256;

    // Degrees once (dst invariant across layers)
    hipMemsetAsync(cnt, 0, (size_t)N_NODES * sizeof(float), stream);
    degree_kernel<<<eblocks, 256, 0, stream>>>(dst, cnt, N_EDGES);
    invdeg_kernel<<<nblocks, 256, 0, stream>>>(cnt, inv, N_NODES);

    const int dins[4]   = {56, 16, 32, 64};
    const int douts[4]  = {16, 32, 64, 2};
    const int doutps[4] = {16, 32, 64, 16};
    float*    houts[4]  = {hA, hB, hA, hB};
    const float* hin = x;

    const int waves       = N_NODES / 16;               // 6250 exactly
    const int gemm_blocks = (waves * 32 + 255) / 256;   // 8 waves / block

    for (int l = 0; l < 4; ++l) {
        const int din = dins[l], dout = douts[l], doutp = doutps[l];

        hipMemsetAsync(agg, 0, (size_t)N_NODES * din * sizeof(float), stream);
        dim3 sgrid(eblocks, din >> 2);
        sage_scatter<<<sgrid, 256, 0, stream>>>(hin, src, dst, agg, N_EDGES, din);

        const int ptotal = doutp * din;
        pack_weights<<<(ptotal + 255) / 256, 256, 0, stream>>>(
            Wls[l], bls[l], Wrs[l], Wlt, Wrt, blp, din, dout, doutp);

        switch (l) {
        case 0:
            sage_fused_gemm<56, 16, 16><<<gemm_blocks, 256, 0, stream>>>(
                hin, agg, inv, Wlt, blp, Wrt, houts[l], N_NODES);
            break;
        case 1:
            sage_fused_gemm<16, 32, 32><<<gemm_blocks, 256, 0, stream>>>(
                hin, agg, inv, Wlt, blp, Wrt, houts[l], N_NODES);
            break;
        case 2:
            sage_fused_gemm<32, 64, 64><<<gemm_blocks, 256, 0, stream>>>(
                hin, agg, inv, Wlt, blp, Wrt, houts[l], N_NODES);
            break;
        default:
            sage_fused_gemm<64, 2, 16><<<gemm_blocks, 256, 0, stream>>>(
                hin, agg, inv, Wlt, blp, Wrt, houts[l], N_NODES);
            break;
        }
        hin = houts[l];
    }

    hipMemsetAsync(pooled, 0, (size_t)N_GRAPHS * 2 * sizeof(float), stream);
    hipMemsetAsync(gcnt,   0, (size_t)N_GRAPHS * sizeof(float), stream);
    pool_kernel<<<nblocks, 256, 0, stream>>>(hin, batch, pooled, gcnt, N_NODES);
    softmax_kernel<<<1, N_GRAPHS, 0, stream>>>(pooled, gcnt, (float*)d_out, N_GRAPHS);
}